// Net_1477468750488
// MI455X (gfx1250) — compile-verified
//
#include <hip/hip_runtime.h>
#include <hip/hip_bf16.h>

#define N_NODES  25000
#define N_EDGES  50000
#define N_GRAPHS 1024
#define HID      128
#define ET       64            // edges per block tile
#define MS       (ET / 16)     // M-subtiles per wave (B-reuse factor)

typedef __attribute__((ext_vector_type(2))) float v2f;
typedef __attribute__((ext_vector_type(8))) float v8f;

__device__ __forceinline__ float elu1(float v) {
    return v > 0.0f ? v : (__expf(v) - 1.0f);
}

// -------------------------------------------------------------------------
// Stage 1: h = relu(edge_attr @ w1 + b1)   [N_EDGES x HID], w1 is [5 x HID]
// -------------------------------------------------------------------------
__global__ void edge_mlp_kernel(const float* __restrict__ edge_attr,
                                const float* __restrict__ w1,
                                const float* __restrict__ b1,
                                float* __restrict__ h, int n_edges) {
    int t = blockIdx.x * blockDim.x + threadIdx.x;
    if (t >= n_edges * HID) return;
    int e = t / HID;
    int j = t - e * HID;
    const float* ea = edge_attr + (size_t)e * 5;
    float s = b1[j];
    #pragma unroll
    for (int d = 0; d < 5; ++d) s += ea[d] * w1[d * HID + j];
    h[t] = s > 0.0f ? s : 0.0f;
}

// -------------------------------------------------------------------------
// Stage 2: fused per-edge message GEMM via fp32 WMMA.
//   msg[e,o] = sum_{i,k} (x[src_e,i]*h[e,k]) * w2[k, i*M_OUT+o]  (+ x·b2 term)
// Block = 64-edge tile; one wave per 16-wide N tile; each wave keeps 4
// accumulators (4 M-subtiles) so each B fragment is reused 4x from registers.
// -------------------------------------------------------------------------
template <int M_IN, int M_OUT>
__global__ __launch_bounds__(32 * (M_OUT / 16))
void nnconv_msg_kernel(const float* __restrict__ x,       // [N_NODES, M_IN]
                       const float* __restrict__ h,       // [N_EDGES, HID]
                       const float* __restrict__ w2,      // [HID, M_IN*M_OUT]
                       const float* __restrict__ b2,      // [M_IN*M_OUT]
                       const long long* __restrict__ src,
                       const long long* __restrict__ dst,
                       float* __restrict__ agg,           // [N_NODES, M_OUT]
                       int n_edges) {
    __shared__ float x_s[ET][M_IN + 1];   // padded: conflict-free column reads
    __shared__ float h_s[ET][HID + 1];

    const int tile = blockIdx.x * ET;
    const int tid  = threadIdx.x;
    const int nthr = blockDim.x;

    // cooperative gather of x[src] rows for this tile
    for (int t = tid; t < ET * M_IN; t += nthr) {
        int m = t / M_IN, i = t - m * M_IN;
        int e = tile + m;
        float v = 0.0f;
        if (e < n_edges) v = x[(size_t)src[e] * M_IN + i];
        x_s[m][i] = v;
    }
    // cooperative load of h rows (zero-padded past n_edges so WMMA adds 0)
    for (int t = tid; t < ET * HID; t += nthr) {
        int m = t >> 7, k = t & (HID - 1);
        int e = tile + m;
        h_s[m][k] = (e < n_edges) ? h[(size_t)e * HID + k] : 0.0f;
    }
    __syncthreads();

    const int wave   = tid >> 5;          // N-tile index
    const int lane   = tid & 31;
    const int row    = lane & 15;         // A: M row / B,C: N column (lane-local)
    const int kb     = (lane >> 4) << 1;  // K sub-offset: lanes 0-15 -> {0,1}, 16-31 -> {2,3}
    const int n_base = wave * 16;
    const size_t kstride = (size_t)M_IN * M_OUT;   // w2 row stride

    v8f acc[MS];
    #pragma unroll
    for (int ms = 0; ms < MS; ++ms)
        acc[ms] = (v8f){0.f, 0.f, 0.f, 0.f, 0.f, 0.f, 0.f, 0.f};

    for (int i = 0; i < M_IN; ++i) {
        float xv[MS];                                       // A scalars, reused over k
        #pragma unroll
        for (int ms = 0; ms < MS; ++ms) xv[ms] = x_s[ms * 16 + row][i];
        const float* w2col = w2 + (size_t)i * M_OUT + n_base + row;
        #pragma unroll 4
        for (int k4 = 0; k4 < HID; k4 += 4) {
            v2f b;
            b[0] = w2col[(size_t)(k4 + kb)     * kstride];  // L2-resident B,
            b[1] = w2col[(size_t)(k4 + kb + 1) * kstride];  // reused by MS subtiles
            #pragma unroll
            for (int ms = 0; ms < MS; ++ms) {
                v2f a;
                a[0] = xv[ms] * h_s[ms * 16 + row][k4 + kb];
                a[1] = xv[ms] * h_s[ms * 16 + row][k4 + kb + 1];
                acc[ms] = __builtin_amdgcn_wmma_f32_16x16x4_f32(
                        /*neg_a=*/false, a, /*neg_b=*/false, b,
                        /*c_mod=*/(short)0, acc[ms],
                        /*reuse_a=*/false, /*reuse_b=*/false);
            }
        }
    }

    // C layout: VGPR r -> M = r (lanes 0-15) / r+8 (lanes 16-31); N = lane&15
    const int mbase = (lane < 16) ? 0 : 8;
    const int ocol  = n_base + row;
    #pragma unroll
    for (int ms = 0; ms < MS; ++ms) {
        #pragma unroll
        for (int r = 0; r < 8; ++r) {
            int m = ms * 16 + mbase + r;
            int e = tile + m;
            if (e < n_edges) {
                // + x[src]·b2 term (We bias folded through the edge matmul)
                float bt = 0.0f;
                for (int i = 0; i < M_IN; ++i)
                    bt += x_s[m][i] * b2[i * M_OUT + ocol];
                long long d = dst[e];
                atomicAdd(&agg[(size_t)d * M_OUT + ocol], acc[ms][r] + bt);
            }
        }
    }
}

// -------------------------------------------------------------------------
// Stage 3: x_out = elu(x_in @ root + agg + bias)
// -------------------------------------------------------------------------
template <int M_IN, int M_OUT>
__global__ void node_update_kernel(const float* __restrict__ x_in,
                                   const float* __restrict__ root,
                                   const float* __restrict__ bias,
                                   const float* __restrict__ agg,
                                   float* __restrict__ x_out, int n_nodes) {
    int t = blockIdx.x * blockDim.x + threadIdx.x;
    if (t >= n_nodes * M_OUT) return;
    int v = t / M_OUT;
    int o = t - v * M_OUT;
    const float* xv = x_in + (size_t)v * M_IN;
    float s = bias[o] + agg[t];
    #pragma unroll 4
    for (int i = 0; i < M_IN; ++i) s += xv[i] * root[i * M_OUT + o];
    x_out[t] = elu1(s);
}

// -------------------------------------------------------------------------
// Pooling: segment-sum of node features + counts per graph
// -------------------------------------------------------------------------
__global__ void pool_kernel(const float* __restrict__ x,
                            const long long* __restrict__ batch,
                            float* __restrict__ sums, float* __restrict__ cnt,
                            int n_nodes) {
    int t = blockIdx.x * blockDim.x + threadIdx.x;
    if (t >= n_nodes * 64) return;
    int v = t >> 6;
    int o = t & 63;
    long long g = batch[v];
    atomicAdd(&sums[(size_t)g * 64 + o], x[t]);
    if (o == 0) atomicAdd(&cnt[g], 1.0f);
}

// -------------------------------------------------------------------------
// Readout MLP: g -> elu(fc1) -> elu(fc2) -> fc3. One thread per graph.
// -------------------------------------------------------------------------
__global__ void readout_kernel(const float* __restrict__ sums,
                               const float* __restrict__ cnt,
                               const float* __restrict__ fc1_w, const float* __restrict__ fc1_b,
                               const float* __restrict__ fc2_w, const float* __restrict__ fc2_b,
                               const float* __restrict__ fc3_w, const float* __restrict__ fc3_b,
                               float* __restrict__ out) {
    int g = blockIdx.x * blockDim.x + threadIdx.x;
    if (g >= N_GRAPHS) return;
    float inv = 1.0f / fmaxf(cnt[g], 1.0f);
    float feat[64];
    #pragma unroll 8
    for (int o = 0; o < 64; ++o) feat[o] = sums[(size_t)g * 64 + o] * inv;
    float a1[32];
    for (int j = 0; j < 32; ++j) {
        float s = fc1_b[j];
        #pragma unroll 8
        for (int o = 0; o < 64; ++o) s += feat[o] * fc1_w[o * 32 + j];
        a1[j] = elu1(s);
    }
    float a2[16];
    for (int j = 0; j < 16; ++j) {
        float s = fc2_b[j];
        #pragma unroll 8
        for (int o = 0; o < 32; ++o) s += a1[o] * fc2_w[o * 16 + j];
        a2[j] = elu1(s);
    }
    float s = fc3_b[0];
    #pragma unroll
    for (int j = 0; j < 16; ++j) s += a2[j] * fc3_w[j];
    out[g] = s;
}

// -------------------------------------------------------------------------
// Host-side orchestration
// -------------------------------------------------------------------------
extern "C" void kernel_launch(void* const* d_in, const int* in_sizes, int n_in,
                              void* d_out, int out_size, void* d_ws, size_t ws_size,
                              hipStream_t stream) {
    (void)in_sizes; (void)n_in; (void)out_size; (void)ws_size;

    const float*     x0   = (const float*)d_in[0];
    const long long* eidx = (const long long*)d_in[1];
    const float*     ea   = (const float*)d_in[2];
    const long long* batch= (const long long*)d_in[3];
    const long long* src  = eidx;
    const long long* dst  = eidx + N_EDGES;

    // per-layer params start at 4 + 6*(layer-1)
    const float* c1[6]; const float* c2[6]; const float* c3[6];
    for (int j = 0; j < 6; ++j) { c1[j] = (const float*)d_in[4 + j];
                                  c2[j] = (const float*)d_in[10 + j];
                                  c3[j] = (const float*)d_in[16 + j]; }
    const float* fc1_w = (const float*)d_in[22]; const float* fc1_b = (const float*)d_in[23];
    const float* fc2_w = (const float*)d_in[24]; const float* fc2_b = (const float*)d_in[25];
    const float* fc3_w = (const float*)d_in[26]; const float* fc3_b = (const float*)d_in[27];
    float* out = (float*)d_out;

    // workspace layout (floats)
    float* ws   = (float*)d_ws;
    float* h    = ws;                                   // 50000*128
    float* bufA = h    + (size_t)N_EDGES * HID;         // 25000*64
    float* bufB = bufA + (size_t)N_NODES * 64;          // 25000*64
    float* agg  = bufB + (size_t)N_NODES * 64;          // 25000*64
    float* sums = agg  + (size_t)N_NODES * 64;          // 1024*64
    float* cnt  = sums + (size_t)N_GRAPHS * 64;         // 1024

    const int EMLP_BLK = 256;
    const int emlp_grid = (N_EDGES * HID + EMLP_BLK - 1) / EMLP_BLK;
    const int msg_grid  = (N_EDGES + ET - 1) / ET;

    // ---- Layer 1: 13 -> 32 ----
    edge_mlp_kernel<<<emlp_grid, EMLP_BLK, 0, stream>>>(ea, c1[0], c1[1], h, N_EDGES);
    hipMemsetAsync(agg, 0, (size_t)N_NODES * 32 * sizeof(float), stream);
    nnconv_msg_kernel<13, 32><<<msg_grid, 64, 0, stream>>>(x0, h, c1[2], c1[3], src, dst, agg, N_EDGES);
    node_update_kernel<13, 32><<<(N_NODES * 32 + 255) / 256, 256, 0, stream>>>(x0, c1[4], c1[5], agg, bufA, N_NODES);

    // ---- Layer 2: 32 -> 64 ----
    edge_mlp_kernel<<<emlp_grid, EMLP_BLK, 0, stream>>>(ea, c2[0], c2[1], h, N_EDGES);
    hipMemsetAsync(agg, 0, (size_t)N_NODES * 64 * sizeof(float), stream);
    nnconv_msg_kernel<32, 64><<<msg_grid, 128, 0, stream>>>(bufA, h, c2[2], c2[3], src, dst, agg, N_EDGES);
    node_update_kernel<32, 64><<<(N_NODES * 64 + 255) / 256, 256, 0, stream>>>(bufA, c2[4], c2[5], agg, bufB, N_NODES);

    // ---- Layer 3: 64 -> 64 ----
    edge_mlp_kernel<<<emlp_grid, EMLP_BLK, 0, stream>>>(ea, c3[0], c3[1], h, N_EDGES);
    hipMemsetAsync(agg, 0, (size_t)N_NODES * 64 * sizeof(float), stream);
    nnconv_msg_kernel<64, 64><<<msg_grid, 128, 0, stream>>>(bufB, h, c3[2], c3[3], src, dst, agg, N_EDGES);
    node_update_kernel<64, 64><<<(N_NODES * 64 + 255) / 256, 256, 0, stream>>>(bufB, c3[4], c3[5], agg, bufA, N_NODES);

    // ---- Pool + readout ----
    hipMemsetAsync(sums, 0, ((size_t)N_GRAPHS * 64 + N_GRAPHS) * sizeof(float), stream);
    pool_kernel<<<(N_NODES * 64 + 255) / 256, 256, 0, stream>>>(bufA, batch, sums, cnt, N_NODES);
    readout_kernel<<<(N_GRAPHS + 255) / 256, 256, 0, stream>>>(sums, cnt, fc1_w, fc1_b, fc2_w, fc2_b, fc3_w, fc3_b, out);
}